// HAN_32435593019723
// MI455X (gfx1250) — compile-verified
//
#include <hip/hip_runtime.h>
#include <math.h>

// ---------------- problem constants ----------------
constexpr int EN   = 16384;   // experts
constexpr int CH   = 256;     // channels
constexpr int HD   = 8;       // heads
constexpr float NEG_SLOPE = 0.2f;

typedef float v2f __attribute__((ext_vector_type(2)));
typedef float v8f __attribute__((ext_vector_type(8)));
typedef unsigned int v4u __attribute__((ext_vector_type(4)));
typedef unsigned int v8u __attribute__((ext_vector_type(8)));

// LDS leading dim for staged A (stride 260: 260%64==4 -> 16 consecutive rows
// hit 16 distinct banks; even stride keeps float2 loads 8B aligned).
// TDM padding reproduces this: pad_interval code 7 (256 dwords) + pad_amount
// code 3 (4 dwords) => 260-dword effective row stride in LDS.
constexpr int LDA = 260;
constexpr int GEMM_SH_BYTES  = 64 * LDA * 4;                 // 66,560 B
constexpr int GEMM2_SH_BYTES = (2 * 64 * LDA + CH) * 4;      // 134,144 B

__device__ __forceinline__ float lrelu(float v) {
  return v > 0.f ? v : NEG_SLOPE * v;
}
__device__ __forceinline__ float waveReduceSum(float v) {
  #pragma unroll
  for (int off = 16; off > 0; off >>= 1) v += __shfl_xor(v, off, 32);
  return v;
}

// ---------------- tiny utility kernels ----------------
__global__ void k_zero(float* p, int n) {
  int i = blockIdx.x * blockDim.x + threadIdx.x;
  if (i < n) p[i] = 0.f;
}

// h_Arrived partial: x[0:49152] @ W_a (49152x256), chunked over K, atomic acc
__global__ void k_harr_partial(const float* __restrict__ x,
                               const float* __restrict__ Wa,
                               float* __restrict__ acc) {
  int c  = threadIdx.x;
  int k0 = blockIdx.x * 256;
  float s = 0.f;
  for (int k = 0; k < 256; ++k)
    s += x[k0 + k] * Wa[(size_t)(k0 + k) * CH + c];
  atomicAdd(acc + c, s);
}

// finalize h_Arrived: +bias, relu row (out_ae constant row), and per-head
// dst-attention dot for edge 'ea' (a_d at Arrived node 0)
__global__ void k_harr_finalize(const float* __restrict__ acc,
                                const float* __restrict__ ba,
                                const float* __restrict__ attD,
                                float* __restrict__ hArr,
                                float* __restrict__ aeRow,
                                float* __restrict__ ad) {
  int c = threadIdx.x;
  float v = acc[c] + ba[c];
  hArr[c]  = v;
  aeRow[c] = fmaxf(v, 0.f);
  float t = waveReduceSum(v * attD[c]);
  if ((c & 31) == 0) ad[c >> 5] = t;
}

// h_Expert = feats(16384x3) @ W(3x256) + b
__global__ void k_hexp(const float* __restrict__ x,
                       const float* __restrict__ We,
                       const float* __restrict__ be,
                       float* __restrict__ hExp) {
  int e = blockIdx.x, c = threadIdx.x;
  const float* f = x + 3 * EN + (size_t)e * 123;
  float v = be[c] + f[0] * We[c] + f[1] * We[CH + c] + f[2] * We[2 * CH + c];
  hExp[(size_t)e * CH + c] = v;
}

// Fused 're'/'we': project 10 children (K=6) in-registers, per-head GAT
// softmax over the 10 edges, aggregate, relu. One block per expert.
__global__ void k_child_edge(const float* __restrict__ x, int featOff,
                             const float* __restrict__ W6,
                             const float* __restrict__ b6,
                             const float* __restrict__ attS,
                             const float* __restrict__ attD,
                             const float* __restrict__ hExp,
                             float* __restrict__ outBuf) {
  __shared__ float Wl[6 * CH];
  __shared__ float feats[60];
  int e = blockIdx.x, c = threadIdx.x;
  #pragma unroll
  for (int k = 0; k < 6; ++k) Wl[k * CH + c] = W6[k * CH + c];
  if (c < 60) feats[c] = x[3 * EN + (size_t)e * 123 + featOff + c];
  __syncthreads();

  float ad   = waveReduceSum(hExp[(size_t)e * CH + c] * attD[c]);
  float attv = attS[c];
  float bv   = b6[c];
  float hj[10], alpha[10];
  #pragma unroll
  for (int j = 0; j < 10; ++j) {
    float hv = bv;
    #pragma unroll
    for (int k = 0; k < 6; ++k) hv += feats[j * 6 + k] * Wl[k * CH + c];
    hj[j] = hv;
    float as = waveReduceSum(hv * attv);
    alpha[j] = lrelu(as + ad);
  }
  float m = alpha[0];
  #pragma unroll
  for (int j = 1; j < 10; ++j) m = fmaxf(m, alpha[j]);
  float s = 0.f, o = 0.f;
  #pragma unroll
  for (int j = 0; j < 10; ++j) {
    float w = expf(alpha[j] - m);
    s += w; o += w * hj[j];
  }
  outBuf[(size_t)e * CH + c] = fmaxf(o / s, 0.f);
}

// 'ea' edge softmax over 16384 experts: pass 1 = alpha + per-block max
__global__ void k_ea_alpha(const float* __restrict__ hsrc,
                           const float* __restrict__ attS,
                           const float* __restrict__ ad,
                           float* __restrict__ alpha,
                           float* __restrict__ maxPart) {
  int c = threadIdx.x, wv = c >> 5;
  float attv = attS[c];
  float adv  = ad[wv];
  float m = -3.4e38f;
  int e0 = blockIdx.x * 32;
  for (int i = 0; i < 32; ++i) {
    int e = e0 + i;
    float a = waveReduceSum(hsrc[(size_t)e * CH + c] * attv);
    a = lrelu(a + adv);
    if ((c & 31) == 0) alpha[e * HD + wv] = a;
    m = fmaxf(m, a);
  }
  if ((c & 31) == 0) maxPart[blockIdx.x * HD + wv] = m;
}

__global__ void k_max_reduce(const float* __restrict__ maxPart,
                             float* __restrict__ amax, int nb) {
  int h = threadIdx.x;
  float m = -3.4e38f;
  for (int g = 0; g < nb; ++g) m = fmaxf(m, maxPart[g * HD + h]);
  amax[h] = m;
}

// pass 2: exp-sum + weighted aggregation of h_Expert rows
__global__ void k_ea_sum_agg(const float* __restrict__ hsrc,
                             const float* __restrict__ alpha,
                             const float* __restrict__ amax,
                             float* __restrict__ agg,
                             float* __restrict__ denom) {
  int c = threadIdx.x, wv = c >> 5;
  float amx = amax[wv];
  float accv = 0.f, dsum = 0.f;
  int e0 = blockIdx.x * 32;
  for (int i = 0; i < 32; ++i) {
    int e = e0 + i;
    float w = expf(alpha[e * HD + wv] - amx);
    accv += w * hsrc[(size_t)e * CH + c];
    dsum += w;
  }
  atomicAdd(agg + c, accv);
  if ((c & 31) == 0) atomicAdd(denom + wv, dsum);
}

__global__ void k_ea_finalize(const float* __restrict__ agg,
                              const float* __restrict__ denom,
                              float* __restrict__ outv) {
  int c = threadIdx.x;
  outv[c] = fmaxf(agg[c] / denom[c >> 5], 0.f);
}

// colmean of tanh(aeRow @ k_w + k_b) -- out_ae rows are all identical
__global__ void k_colmean_ae(const float* __restrict__ aeRow,
                             const float* __restrict__ kw,
                             const float* __restrict__ kb,
                             float* __restrict__ colmean) {
  __shared__ float r[CH];
  int c = threadIdx.x;
  r[c] = aeRow[c];
  __syncthreads();
  float s = kb[c];
  for (int k = 0; k < CH; ++k) s += r[k] * kw[k * CH + c];
  colmean[c] = tanhf(s);
}

// -------- WMMA f32 GEMM (16x16x4) + tanh + column-sum, TDM-staged A -------
// Block: 512 thr = 16 waves; 64 A-rows DMA'd into LDS by the Tensor Data
// Mover (one descriptor, hardware row padding 256->260 dwords), wave w owns
// a 16-column tile.
__global__ void __launch_bounds__(512) k_gemm_tanh_colsum(
    const float* __restrict__ A, const float* __restrict__ kw,
    const float* __restrict__ kb, float* __restrict__ colsum) {
  extern __shared__ float Alds[];
  const int tid = threadIdx.x;
  const int rowBase = blockIdx.x * 64;
  if (tid < 256) __builtin_prefetch(&kw[tid * CH], 0, 3);

  // ---- TDM: one tensor_load_to_lds per workgroup (wave 0 only; TDM
  // ignores EXEC, so gate with a scalar branch) ----
  if (__builtin_amdgcn_readfirstlane((int)threadIdx.x) < 32) {
    unsigned ldsBase = __builtin_amdgcn_groupstaticsize();
    unsigned long long ga =
        (unsigned long long)(size_t)(A + (size_t)rowBase * CH);
    v4u g0;
    g0[0] = 1u;                                   // count=1, user mode
    g0[1] = ldsBase;                              // lds_addr (bytes)
    g0[2] = (unsigned)(ga & 0xFFFFFFFFu);         // global_addr[95:64]
    g0[3] = (unsigned)((ga >> 32) & 0x01FFFFFFu)  // global_addr[120:96]
            | (2u << 30);                         // type=2 ("image")
    v8u g1;
    g1[0] = (2u << 16)      // data_size = 4B
          | (1u << 20)      // pad_enable
          | (7u << 22)      // pad_interval code 7 = 256 dwords
          | (3u << 25);     // pad_amount  code 3 = 4 dwords
    g1[1] = (256u << 16);   // tensor_dim0 = 256 (low 16 in bits 63:48)
    g1[2] = (16384u << 16); // tensor_dim0 hi = 0; tensor_dim1 lo16 = 16384
    g1[3] = (256u << 16);   // tensor_dim1 hi = 0; tile_dim0 = 256
    g1[4] = 64u;            // tile_dim1 = 64; tile_dim2 = 0
    g1[5] = 256u;           // tensor_dim0_stride (low 32) = 256 elements
    g1[6] = 0u;
    g1[7] = 0u;
    asm volatile("tensor_load_to_lds %0, %1, null, null"
                 :: "s"(g0), "s"(g1) : "memory");
    __builtin_amdgcn_s_wait_tensorcnt(0);
  }
  __syncthreads();

  const int wv = tid >> 5, lane = tid & 31, ln = lane & 15, lk = lane >> 4;
  const int n0 = wv * 16;
  const float bias = kb[n0 + ln];
  float colreg = 0.f;
  for (int ms = 0; ms < 4; ++ms) {
    v8f acc;
    #pragma unroll
    for (int i = 0; i < 8; ++i) acc[i] = bias;
    const float* arow = &Alds[(ms * 16 + ln) * LDA + 2 * lk];
    for (int k = 0; k < CH; k += 4) {
      v2f a = *(const v2f*)(arow + k);           // A: M=ln, K={k+2lk, k+2lk+1}
      v2f b;
      b.x = kw[(k + lk) * CH + n0 + ln];         // B: K rows striped over halves
      b.y = kw[(k + 2 + lk) * CH + n0 + ln];
      acc = __builtin_amdgcn_wmma_f32_16x16x4_f32(
          false, a, false, b, (short)0, acc, false, false);
    }
    float p = 0.f;
    #pragma unroll
    for (int i = 0; i < 8; ++i) p += tanhf(acc[i]);
    p += __shfl_xor(p, 16, 32);   // fold M=r+8 half onto M=r half (same column)
    colreg += p;
  }
  if (lane < 16) atomicAdd(colsum + n0 + lane, colreg);
}

// layer-2 Expert projection with async global->LDS staging:
// out_re/out_we tiles are DMA'd raw (ASYNCcnt-tracked), and the semantic
// combination res1 = a0*aeRow + a1*re + a2*we is fused into the WMMA
// A-fragment load. D = res1 @ W2 + b2, stored.
__global__ void __launch_bounds__(512) k_gemm2(
    const float* __restrict__ outRE, const float* __restrict__ outWE,
    const float* __restrict__ aeRowIn, const float* __restrict__ attn,
    const float* __restrict__ W2, const float* __restrict__ b2,
    float* __restrict__ h2) {
  extern __shared__ float Alds[];
  float* reT = Alds;                 // 64*LDA
  float* weT = Alds + 64 * LDA;      // 64*LDA
  float* aeL = Alds + 2 * 64 * LDA;  // 256
  const int tid = threadIdx.x;
  const int rowBase = blockIdx.x * 64;
  const float a0 = attn[0], a1 = attn[1], a2 = attn[2];
  if (tid < 256) __builtin_prefetch(&W2[tid * CH], 0, 3);

  // ---- async global->LDS copies of the two source tiles (16B chunks) ----
  const float* reSrc = outRE + (size_t)rowBase * CH;
  const float* weSrc = outWE + (size_t)rowBase * CH;
  const unsigned ldsBase = __builtin_amdgcn_groupstaticsize();
  const unsigned weOfs   = (unsigned)(64 * LDA * 4);
  for (int ci = tid; ci < 64 * 64; ci += 512) {
    int rr = ci >> 6;                 // row 0..63
    int c4 = (ci & 63) << 2;          // column (floats), multiple of 4
    unsigned lo = ldsBase + (unsigned)((rr * LDA + c4) * 4);
    unsigned go = (unsigned)((rr * CH + c4) * 4);
    asm volatile("global_load_async_to_lds_b128 %0, %1, %2"
                 :: "v"(lo), "v"(go), "s"(reSrc) : "memory");
    asm volatile("global_load_async_to_lds_b128 %0, %1, %2"
                 :: "v"(lo + weOfs), "v"(go), "s"(weSrc) : "memory");
  }
  if (tid < CH) aeL[tid] = aeRowIn[tid];
  asm volatile("s_wait_asynccnt 0x0" ::: "memory");
  __syncthreads();

  const int wv = tid >> 5, lane = tid & 31, ln = lane & 15, lk = lane >> 4;
  const int n0 = wv * 16;
  const float bias = b2[n0 + ln];
  for (int ms = 0; ms < 4; ++ms) {
    v8f acc;
    #pragma unroll
    for (int i = 0; i < 8; ++i) acc[i] = bias;
    const float* reRow = &reT[(ms * 16 + ln) * LDA + 2 * lk];
    const float* weRow = &weT[(ms * 16 + ln) * LDA + 2 * lk];
    for (int k = 0; k < CH; k += 4) {
      v2f re = *(const v2f*)(reRow + k);
      v2f we = *(const v2f*)(weRow + k);
      v2f ar = *(const v2f*)(aeL + k + 2 * lk);  // indexed by K (= channel)
      v2f a;
      a.x = a0 * ar.x + a1 * re.x + a2 * we.x;   // fused res1 construction
      a.y = a0 * ar.y + a1 * re.y + a2 * we.y;
      v2f b;
      b.x = W2[(k + lk) * CH + n0 + ln];
      b.y = W2[(k + 2 + lk) * CH + n0 + ln];
      acc = __builtin_amdgcn_wmma_f32_16x16x4_f32(
          false, a, false, b, (short)0, acc, false, false);
    }
    #pragma unroll
    for (int i = 0; i < 8; ++i) {       // C layout: lanes 0-15 M=i, 16-31 M=i+8
      int row = rowBase + ms * 16 + i + 8 * lk;
      h2[(size_t)row * CH + n0 + ln] = acc[i];
    }
  }
}

// semantic scores for Expert (3-way softmax over {ae, re, we})
__global__ void k_scores(const float* __restrict__ cmAE,
                         const float* __restrict__ csRE,
                         const float* __restrict__ csWE,
                         const float* __restrict__ q,
                         float* __restrict__ attn) {
  __shared__ float s0[CH], s1[CH], s2[CH];
  int c = threadIdx.x;
  float qv = q[c];
  s0[c] = qv * cmAE[c];
  s1[c] = qv * (csRE[c] * (1.f / (float)EN));
  s2[c] = qv * (csWE[c] * (1.f / (float)EN));
  __syncthreads();
  for (int off = 128; off > 0; off >>= 1) {
    if (c < off) { s0[c] += s0[c + off]; s1[c] += s1[c + off]; s2[c] += s2[c + off]; }
    __syncthreads();
  }
  if (c == 0) {
    float x0 = s0[0], x1 = s1[0], x2 = s2[0];
    float m  = fmaxf(x0, fmaxf(x1, x2));
    float e0 = expf(x0 - m), e1 = expf(x1 - m), e2 = expf(x2 - m);
    float inv = 1.f / (e0 + e1 + e2);
    attn[0] = e0 * inv; attn[1] = e1 * inv; attn[2] = e2 * inv;
  }
}

// layer-2 Arrived projection (256x256 matvec) + 'ea' dst-attention dot
__global__ void k_matvec_ad(const float* __restrict__ vin,
                            const float* __restrict__ W,
                            const float* __restrict__ b,
                            const float* __restrict__ attD,
                            float* __restrict__ vout,
                            float* __restrict__ ad) {
  __shared__ float r[CH];
  int c = threadIdx.x;
  r[c] = vin[c];
  __syncthreads();
  float s = b[c];
  for (int k = 0; k < CH; ++k) s += r[k] * W[k * CH + c];
  vout[c] = s;
  float t = waveReduceSum(s * attD[c]);
  if ((c & 31) == 0) ad[c >> 5] = t;
}

// ---------------- host-side orchestration ----------------
extern "C" void kernel_launch(void* const* d_in, const int* in_sizes, int n_in,
                              void* d_out, int out_size, void* d_ws, size_t ws_size,
                              hipStream_t stream) {
  (void)in_sizes; (void)n_in; (void)out_size; (void)ws_size;
  const float* x = (const float*)d_in[0];
  // params flattened in dict-insertion order:
  //  +0 Arr.w +1 Arr.b +2 Exp.w +3 Exp.b +4 Run.w +5 Run.b +6 Wait.w +7 Wait.b
  //  +8 ae.src +9 ae.dst +10 ea.src +11 ea.dst +12 er.src +13 er.dst
  //  +14 re.src +15 re.dst +16 ew.src +17 ew.dst +18 we.src +19 we.dst
  //  +20 k_w +21 k_b +22 q
  #define P1(i) ((const float*)d_in[1  + (i)])
  #define P2(i) ((const float*)d_in[24 + (i)])

  float* ws     = (float*)d_ws;
  float* hExp   = ws;                                   // EN*CH (later: h2Exp)
  float* outRE  = ws + (size_t)EN * CH;                 // EN*CH
  float* outWE  = ws + (size_t)2 * EN * CH;             // EN*CH
  float* alpha  = ws + (size_t)3 * EN * CH;             // EN*HD
  float* sm     = alpha + (size_t)EN * HD;              // small scratch region
  float* hArrAcc = sm;          // 256
  float* hArr    = sm + 256;    // 256
  float* aeRow   = sm + 512;    // 256
  float* adBuf   = sm + 768;    // 8
  float* amax    = sm + 776;    // 8
  float* denom   = sm + 784;    // 8
  float* agg     = sm + 792;    // 256 (contiguous with denom for re-zero)
  float* resArr  = sm + 1048;   // 256
  float* h2Arr   = sm + 1304;   // 256
  float* cmAE    = sm + 1560;   // 256
  float* csRE    = sm + 1816;   // 256
  float* csWE    = sm + 2072;   // 256
  float* attn    = sm + 2328;   // 3
  float* maxPart = sm + 2336;   // 512*8

  // ---- layer 1 ----
  k_zero<<<32, 256, 0, stream>>>(sm, 8192);
  k_harr_partial<<<192, 256, 0, stream>>>(x, P1(0), hArrAcc);
  k_harr_finalize<<<1, 256, 0, stream>>>(hArrAcc, P1(1), P1(11), hArr, aeRow, adBuf);
  k_hexp<<<EN, 256, 0, stream>>>(x, P1(2), P1(3), hExp);
  k_child_edge<<<EN, 256, 0, stream>>>(x, 3,  P1(4), P1(5), P1(14), P1(15), hExp, outRE);
  k_child_edge<<<EN, 256, 0, stream>>>(x, 63, P1(6), P1(7), P1(18), P1(19), hExp, outWE);
  k_ea_alpha<<<512, 256, 0, stream>>>(hExp, P1(10), adBuf, alpha, maxPart);
  k_max_reduce<<<1, 8, 0, stream>>>(maxPart, amax, 512);
  k_ea_sum_agg<<<512, 256, 0, stream>>>(hExp, alpha, amax, agg, denom);
  k_ea_finalize<<<1, 256, 0, stream>>>(agg, denom, resArr);       // res1_Arrived

  // ---- semantic attention for Expert (WMMA GEMMs, TDM-staged) ----
  k_colmean_ae<<<1, 256, 0, stream>>>(aeRow, P1(20), P1(21), cmAE);
  k_gemm_tanh_colsum<<<256, 512, GEMM_SH_BYTES, stream>>>(outRE, P1(20), P1(21), csRE);
  k_gemm_tanh_colsum<<<256, 512, GEMM_SH_BYTES, stream>>>(outWE, P1(20), P1(21), csWE);
  k_scores<<<1, 256, 0, stream>>>(cmAE, csRE, csWE, P1(22), attn);

  // ---- layer 2 (only Arrived path matters) ----
  k_gemm2<<<256, 512, GEMM2_SH_BYTES, stream>>>(outRE, outWE, aeRow, attn,
                                                P2(2), P2(3), hExp /* h2Exp */);
  k_matvec_ad<<<1, 256, 0, stream>>>(resArr, P2(0), P2(1), P2(11), h2Arr, adBuf);
  k_zero<<<2, 256, 0, stream>>>(denom, 264);   // re-zero denom+agg
  k_ea_alpha<<<512, 256, 0, stream>>>(hExp, P2(10), adBuf, alpha, maxPart);
  k_max_reduce<<<1, 8, 0, stream>>>(maxPart, amax, 512);
  k_ea_sum_agg<<<512, 256, 0, stream>>>(hExp, alpha, amax, agg, denom);
  k_ea_finalize<<<1, 256, 0, stream>>>(agg, denom, (float*)d_out);

  #undef P1
  #undef P2
}